// CausalAttention_73212012527759
// MI455X (gfx1250) — compile-verified
//
#include <hip/hip_runtime.h>

// MI455X / gfx1250, wave32. All matrix math on V_WMMA_F32_16X16X4_F32.
typedef __attribute__((ext_vector_type(2))) float v2f;
typedef __attribute__((ext_vector_type(8))) float v8f;

#define PI_F 3.14159265358979323846f

static __device__ __forceinline__ v8f wmma4(v2f a, v2f b, v8f c) {
  // 8 args: (neg_a, A, neg_b, B, c_mod, C, reuse_a, reuse_b)
  return __builtin_amdgcn_wmma_f32_16x16x4_f32(false, a, false, b, (short)0, c,
                                               false, false);
}

// Dimensions fixed by setup_inputs(): B=4, L=2048, H=16, E=D=64.
#define NPOS   8192          // B*L
#define WAVES  8             // waves per block
#define ORDER_W 0.2f

__global__ __launch_bounds__(256) void CausalAttention_fft_wmma(
    const float* __restrict__ Q, const float* __restrict__ K,
    const float* __restrict__ V, float* __restrict__ Out) {
  // ---------------- constant tables (per block, LDS) ----------------
  __shared__ float CB[64][16];    // cos(2*pi*e*k/64)  k<9, else 0
  __shared__ float SB[64][16];    // -sin(2*pi*e*k/64) k<9, else 0
  __shared__ float cosT[16][16];  // cos(2*pi*r*c/16)
  __shared__ float psinT[16][16]; // +sin
  __shared__ float nsinT[16][16]; // -sin
  __shared__ float Gc[16][16];    // (ck/256)*cos(2*pi*k*n/16), k<9 else 0
  __shared__ float Gs[16][16];    // -(ck/256)*sin(...),        k<9 else 0
  __shared__ float Wc[16][16];    // causal cumprod weights
  __shared__ float scr[WAVES][2][16][17]; // per-wave layout-shuffle scratch

  const int tid = threadIdx.x;
  for (int i = tid; i < 64 * 16; i += 256) {
    int e = i >> 4, k = i & 15;
    float c = 0.f, s = 0.f;
    if (k < 9) {
      float ang = (PI_F / 32.0f) * (float)(e * k); // 2*pi*e*k/64
      c = cosf(ang);
      s = -sinf(ang);
    }
    CB[e][k] = c;
    SB[e][k] = s;
  }
  {
    int r = tid >> 4, c = tid & 15;
    float ang = (PI_F / 8.0f) * (float)(r * c); // 2*pi*r*c/16
    float cv = cosf(ang), sv = sinf(ang);
    cosT[r][c] = cv;
    psinT[r][c] = sv;
    nsinT[r][c] = -sv;
    float gc = 0.f, gs = 0.f;
    if (r < 9) {
      float ck = (r == 0 || r == 8) ? 1.0f : 2.0f;
      gc = (ck * (1.0f / 256.0f)) * cv;
      gs = -(ck * (1.0f / 256.0f)) * sv;
    }
    Gc[r][c] = gc;
    Gs[r][c] = gs;
  }
  if (tid == 0) {
    float w = 1.0f;
    Wc[0][0] = 1.0f;
    for (int j = 1; j < 256; ++j) {
      w *= fabsf(1.0f - (ORDER_W + 1.0f) / (float)j);
      Wc[j >> 4][j & 15] = w;
    }
  }
  __syncthreads();

  // ---------------- per-wave: one (b,l) position ----------------
  const int wave = tid >> 5;
  const int lane = tid & 31;
  const int col = lane & 15;  // N-index / A-row (M) index
  const int hi = lane >> 4;   // lane-half select
  const int pos = blockIdx.x * WAVES + wave; // 0..8191

  const float* q = Q + (size_t)pos * 1024;
  const float* k = K + (size_t)pos * 1024;
  const float* v = V + (size_t)pos * 1024;
  float* o = Out + (size_t)pos * 1024;

  // Prefetch V (16x64 f32 = 4KB = 32 x 128B lines) while transforms run.
  __builtin_prefetch(v + lane * 32, 0, 1);

  float(*s0)[17] = scr[wave][0];
  float(*s1)[17] = scr[wave][1];

  // ---- forward along e:  Zq = Q @ (C | S)   (16x64 @ 64x16, K=64) ----
  v8f zc = {}, zs = {};
#pragma unroll
  for (int t = 0; t < 16; ++t) {
    int e0 = 4 * t + 2 * hi;
    v2f a = {q[col * 64 + e0], q[col * 64 + e0 + 1]};
    v2f bc = {CB[e0][col], CB[e0 + 1][col]};
    v2f bs = {SB[e0][col], SB[e0 + 1][col]};
    zc = wmma4(a, bc, zc);
    zs = wmma4(a, bs, zs);
  }
  // D-layout -> LDS (plain [row][col]) so Z can be reused as a B operand.
#pragma unroll
  for (int r = 0; r < 8; ++r) {
    s0[r + 8 * hi][col] = zc[r];
    s1[r + 8 * hi][col] = zs[r];
  }
  // ---- forward along h:  Wq = F16 @ Zq  (complex; F16 = cos - i*sin) ----
  v8f wqr = {}, wqi = {};
#pragma unroll
  for (int t = 0; t < 4; ++t) {
    int h0 = 4 * t + 2 * hi;
    v2f ac = {cosT[col][h0], cosT[col][h0 + 1]};
    v2f ap = {psinT[col][h0], psinT[col][h0 + 1]};
    v2f an = {nsinT[col][h0], nsinT[col][h0 + 1]};
    v2f bzc = {s0[h0][col], s0[h0 + 1][col]};
    v2f bzs = {s1[h0][col], s1[h0 + 1][col]};
    wqr = wmma4(ac, bzc, wqr); // Re = cos@Zc + sin@Zs
    wqr = wmma4(ap, bzs, wqr);
    wqi = wmma4(ac, bzs, wqi); // Im = cos@Zs - sin@Zc
    wqi = wmma4(an, bzc, wqi);
  }

  // ---- same two stages for K ----
  zc = (v8f){};
  zs = (v8f){};
#pragma unroll
  for (int t = 0; t < 16; ++t) {
    int e0 = 4 * t + 2 * hi;
    v2f a = {k[col * 64 + e0], k[col * 64 + e0 + 1]};
    v2f bc = {CB[e0][col], CB[e0 + 1][col]};
    v2f bs = {SB[e0][col], SB[e0 + 1][col]};
    zc = wmma4(a, bc, zc);
    zs = wmma4(a, bs, zs);
  }
#pragma unroll
  for (int r = 0; r < 8; ++r) {
    s0[r + 8 * hi][col] = zc[r];
    s1[r + 8 * hi][col] = zs[r];
  }
  v8f wkr = {}, wki = {};
#pragma unroll
  for (int t = 0; t < 4; ++t) {
    int h0 = 4 * t + 2 * hi;
    v2f ac = {cosT[col][h0], cosT[col][h0 + 1]};
    v2f ap = {psinT[col][h0], psinT[col][h0 + 1]};
    v2f an = {nsinT[col][h0], nsinT[col][h0 + 1]};
    v2f bzc = {s0[h0][col], s0[h0 + 1][col]};
    v2f bzs = {s1[h0][col], s1[h0 + 1][col]};
    wkr = wmma4(ac, bzc, wkr);
    wkr = wmma4(ap, bzs, wkr);
    wki = wmma4(ac, bzs, wki);
    wki = wmma4(an, bzc, wki);
  }

  // ---- RES = Wq * conj(Wk)  (elementwise, same D-layout) ----
  v8f rr, ri;
#pragma unroll
  for (int r = 0; r < 8; ++r) {
    rr[r] = wqr[r] * wkr[r] + wqi[r] * wki[r];
    ri[r] = wqi[r] * wkr[r] - wqr[r] * wki[r];
  }
#pragma unroll
  for (int r = 0; r < 8; ++r) {
    s0[r + 8 * hi][col] = rr[r];
    s1[r + 8 * hi][col] = ri[r];
  }

  // ---- inverse along h:  U = iF16 @ RES  (iF16 = cos + i*sin) ----
  v8f ur = {}, ui = {};
#pragma unroll
  for (int t = 0; t < 4; ++t) {
    int j0 = 4 * t + 2 * hi;
    v2f ac = {cosT[col][j0], cosT[col][j0 + 1]};
    v2f ap = {psinT[col][j0], psinT[col][j0 + 1]};
    v2f an = {nsinT[col][j0], nsinT[col][j0 + 1]};
    v2f br = {s0[j0][col], s0[j0 + 1][col]};
    v2f bi = {s1[j0][col], s1[j0 + 1][col]};
    ur = wmma4(ac, br, ur); // Re = cos@Rr - sin@Ri
    ur = wmma4(an, bi, ur);
    ui = wmma4(ac, bi, ui); // Im = cos@Ri + sin@Rr
    ui = wmma4(ap, br, ui);
  }
#pragma unroll
  for (int r = 0; r < 8; ++r) {
    s0[r + 8 * hi][col] = ur[r];
    s1[r + 8 * hi][col] = ui[r];
  }

  // ---- cropped irfft along n:  corr = Ur@Gc + Ui@Gs ----
  v8f corr = {};
#pragma unroll
  for (int t = 0; t < 4; ++t) {
    int k0 = 4 * t + 2 * hi;
    v2f aur = {s0[col][k0], s0[col][k0 + 1]}; // A-layout read
    v2f aui = {s1[col][k0], s1[col][k0 + 1]};
    v2f bgc = {Gc[k0][col], Gc[k0 + 1][col]};
    v2f bgs = {Gs[k0][col], Gs[k0 + 1][col]};
    corr = wmma4(aur, bgc, corr);
    corr = wmma4(aui, bgs, corr);
  }

  // ---- softmax over n (row = one VGPR across 16 lanes), * causal W ----
#pragma unroll
  for (int r = 0; r < 8; ++r) {
    float x = corr[r] * 0.125f; // 1/sqrt(64)
    float mx = x;
    for (int m = 1; m < 16; m <<= 1) mx = fmaxf(mx, __shfl_xor(mx, m, 16));
    float ex = __expf(x - mx);
    float sum = ex;
    for (int m = 1; m < 16; m <<= 1) sum += __shfl_xor(sum, m, 16);
    corr[r] = (ex / sum) * Wc[r + 8 * hi][col];
  }

  // ---- out = corr @ V  (16x16 @ 16x64) ----
#pragma unroll
  for (int r = 0; r < 8; ++r) s0[r + 8 * hi][col] = corr[r];
#pragma unroll
  for (int nt = 0; nt < 4; ++nt) {
    v8f acc = {};
#pragma unroll
    for (int t = 0; t < 4; ++t) {
      int k0 = 4 * t + 2 * hi;
      v2f a = {s0[col][k0], s0[col][k0 + 1]};
      v2f b = {v[k0 * 64 + nt * 16 + col], v[(k0 + 1) * 64 + nt * 16 + col]};
      acc = wmma4(a, b, acc);
    }
#pragma unroll
    for (int r = 0; r < 8; ++r)
      o[(r + 8 * hi) * 64 + nt * 16 + col] = acc[r];
  }
}

extern "C" void kernel_launch(void* const* d_in, const int* in_sizes, int n_in,
                              void* d_out, int out_size, void* d_ws,
                              size_t ws_size, hipStream_t stream) {
  const float* q = (const float*)d_in[0];
  const float* k = (const float*)d_in[1];
  const float* v = (const float*)d_in[2];
  // d_in[3] = attn_mask scalar, unused by the reference math.
  float* out = (float*)d_out;
  dim3 grid(NPOS / WAVES), block(256);
  hipLaunchKernelGGL(CausalAttention_fft_wmma, grid, block, 0, stream, q, k, v,
                     out);
}